// ICDM_EX_71700184039591
// MI455X (gfx1250) — compile-verified
//
#include <hip/hip_runtime.h>

typedef __attribute__((ext_vector_type(2))) float v2f;
typedef __attribute__((ext_vector_type(8))) float v8f;

#define DIM 64

// ---------------- utility kernels ----------------

__global__ __launch_bounds__(256) void sage_zero_kernel(float* __restrict__ p, long n) {
    long i = (long)blockIdx.x * blockDim.x + threadIdx.x;
    if (i < n) p[i] = 0.0f;
}

__global__ __launch_bounds__(256) void sage_degree_kernel(const int* __restrict__ dst,
                                                          float* __restrict__ deg, int E) {
    int e = blockIdx.x * blockDim.x + threadIdx.x;
    if (e < E) atomicAdd(&deg[dst[e]], 1.0f);
}

__global__ __launch_bounds__(256) void sage_invdeg_kernel(float* __restrict__ deg, int n) {
    int i = blockIdx.x * blockDim.x + threadIdx.x;
    if (i < n) deg[i] = 1.0f / fmaxf(deg[i], 1.0f);
}

// 16 lanes per edge: each lane moves a float4 (16*4 = 64 floats = one row).
// Gather h[src] (coalesced 256B per edge) then atomic-add into accum[dst].
// Whole working set is L2-resident (192MB L2 >> 90MB), so atomics resolve in L2.
__global__ __launch_bounds__(256) void sage_scatter_kernel(const float* __restrict__ H,
                                                           const int* __restrict__ src,
                                                           const int* __restrict__ dst,
                                                           float* __restrict__ A, int E) {
    long gid = (long)blockIdx.x * blockDim.x + threadIdx.x;
    int e = (int)(gid >> 4);
    int l = (int)(gid & 15);
    if (e >= E) return;
    int s = src[e];
    int d = dst[e];
    float4 v = ((const float4*)(H + (size_t)s * DIM))[l];
    float* ap = A + (size_t)d * DIM + l * 4;
    atomicAdd(ap + 0, v.x);
    atomicAdd(ap + 1, v.y);
    atomicAdd(ap + 2, v.z);
    atomicAdd(ap + 3, v.w);
}

// ---------------- fused dual-GEMM + epilogue ----------------
// Per wave: one 16-row block. out[16x64] = Hs[16x64] @ Ws + b + (Acc*invdeg)[16x64] @ Wn
// K-loop step 4, four 16-col accumulators -> 8 v_wmma_f32_16x16x4_f32 per k-step,
// 128 WMMA per wave.
// Weights staged in LDS *pair-interleaved*: sW[(k/2)*128 + n*2 + (k&1)], so each
// lane's B-fragment {W[k][n], W[k+1][n]} is one 8B-contiguous ds_load_b64 straight
// into an even-aligned VGPR pair (no v_mov packing before WMMA).
// Epilogue: Tmp = out (optional); Res = (Base ? Base : Res) + out*scale.
__global__ __launch_bounds__(256) void sage_gemm_kernel(
    const float* __restrict__ Hs,      // [N,64] self-path input
    const float* __restrict__ Acc,     // [N,64] neighbor sums
    const float* __restrict__ invdeg,  // [N]
    const float* __restrict__ Ws,      // [64,64] row-major (k,n)
    const float* __restrict__ Wn,      // [64,64]
    const float* __restrict__ bsv,     // [64]
    float* __restrict__ Tmp,           // [N,64] layer output (nullptr to skip)
    float* __restrict__ Res,           // [N,64] running result
    const float* __restrict__ Base,    // nullptr => Res += ; else Res = Base +
    float scale, int nBlocks, int N) {
    __shared__ float sWs[DIM * DIM];
    __shared__ float sWn[DIM * DIM];
    for (int i = threadIdx.x; i < DIM * DIM; i += 256) {
        int k = i >> 6;
        int n = i & 63;
        int il = ((k >> 1) << 7) + (n << 1) + (k & 1);  // pair-interleaved
        sWs[il] = Ws[i];
        sWn[il] = Wn[i];
    }
    __syncthreads();

    int wave = threadIdx.x >> 5;
    int lane = threadIdx.x & 31;
    int rb = blockIdx.x * 8 + wave;
    if (rb >= nBlocks) return;  // uniform per wave -> EXEC all-1s for WMMA

    int row0 = rb * 16;
    int lm = lane & 15;   // N-index within tile / M-index of A row
    int lh = lane >> 4;   // lane-half selector
    int arow = row0 + lm;
    if (arow >= N) arow = N - 1;  // clamp loads; stores predicated below
    float invd = invdeg[arow];
    const float* hrow = Hs + (size_t)arow * DIM;
    const float* crow = Acc + (size_t)arow * DIM;

    v8f acc0 = {}, acc1 = {}, acc2 = {}, acc3 = {};

#pragma unroll
    for (int k0 = 0; k0 < DIM; k0 += 4) {
        int ka = k0 + 2 * lh;  // A frag: lane holds K=ka, ka+1 (ISA 16x4 f32 layout)
        v2f aS = *(const v2f*)(hrow + ka);
        v2f aN = *(const v2f*)(crow + ka);
        aN[0] *= invd;
        aN[1] *= invd;
        // interleaved row base for K-pair (ka, ka+1); ka is always even here
        const float* wsr = sWs + ((ka >> 1) << 7);
        const float* wnr = sWn + ((ka >> 1) << 7);
#define SAGE_TILE(T, ACCV)                                                    \
        {                                                                     \
            int nc2 = ((T) * 16 + lm) * 2;                                    \
            v2f bS = *(const v2f*)(wsr + nc2);                                \
            v2f bN = *(const v2f*)(wnr + nc2);                                \
            ACCV = __builtin_amdgcn_wmma_f32_16x16x4_f32(                     \
                false, aS, false, bS, (short)0, ACCV, false, false);          \
            ACCV = __builtin_amdgcn_wmma_f32_16x16x4_f32(                     \
                false, aN, false, bN, (short)0, ACCV, false, false);          \
        }
        SAGE_TILE(0, acc0)
        SAGE_TILE(1, acc1)
        SAGE_TILE(2, acc2)
        SAGE_TILE(3, acc3)
#undef SAGE_TILE
    }

    // Epilogue. C/D layout: VGPR r -> row = row0 + r + 8*lh, col = tile*16 + lm.
    int mbase = row0 + 8 * lh;
#define SAGE_STORE(T, ACCV)                                                   \
    {                                                                         \
        int nc = (T) * 16 + lm;                                               \
        float bias = bsv[nc];                                                 \
        _Pragma("unroll")                                                     \
        for (int r = 0; r < 8; ++r) {                                         \
            int m = mbase + r;                                                \
            if (m < N) {                                                      \
                float tv = ACCV[r] + bias;                                    \
                size_t idx = (size_t)m * DIM + nc;                            \
                if (Tmp) Tmp[idx] = tv;                                       \
                float base = Base ? Base[idx] : Res[idx];                     \
                Res[idx] = base + tv * scale;                                 \
            }                                                                 \
        }                                                                     \
    }
    SAGE_STORE(0, acc0)
    SAGE_STORE(1, acc1)
    SAGE_STORE(2, acc2)
    SAGE_STORE(3, acc3)
#undef SAGE_STORE
}

// ---------------- launcher ----------------

extern "C" void kernel_launch(void* const* d_in, const int* in_sizes, int n_in,
                              void* d_out, int out_size, void* d_ws, size_t ws_size,
                              hipStream_t stream) {
    const float* h   = (const float*)d_in[0];
    const int* esrc  = (const int*)d_in[1];
    const int* edst  = (const int*)d_in[2];
    const float* Ws0 = (const float*)d_in[3];
    const float* b0  = (const float*)d_in[4];
    const float* Wn0 = (const float*)d_in[5];
    const float* Ws1 = (const float*)d_in[6];
    const float* b1  = (const float*)d_in[7];
    const float* Wn1 = (const float*)d_in[8];
    float* out = (float*)d_out;

    int N = in_sizes[0] / DIM;
    int E = in_sizes[1];

    float* accum = (float*)d_ws;                  // N*64
    float* tmp   = accum + (size_t)N * DIM;       // N*64
    float* deg   = tmp + (size_t)N * DIM;         // N  (becomes invdeg in place)

    long zn = (long)N * DIM;
    long sthreads = (long)E * 16;
    int nBlocks = (N + 15) / 16;
    int gemmGrid = (nBlocks + 7) / 8;

    // degrees (shared by both layers)
    sage_zero_kernel<<<(int)((zn + 255) / 256), 256, 0, stream>>>(accum, zn);
    sage_zero_kernel<<<(N + 255) / 256, 256, 0, stream>>>(deg, N);
    sage_degree_kernel<<<(E + 255) / 256, 256, 0, stream>>>(edst, deg, E);
    sage_invdeg_kernel<<<(N + 255) / 256, 256, 0, stream>>>(deg, N);

    // layer 0: tmp = h@Ws0 + b0 + mean@Wn0 ; out = h + tmp
    sage_scatter_kernel<<<(int)((sthreads + 255) / 256), 256, 0, stream>>>(h, esrc, edst, accum, E);
    sage_gemm_kernel<<<gemmGrid, 256, 0, stream>>>(h, accum, deg, Ws0, Wn0, b0,
                                                   tmp, out, h, 1.0f, nBlocks, N);

    // layer 1: out += 0.5 * (tmp@Ws1 + b1 + mean@Wn1)
    sage_zero_kernel<<<(int)((zn + 255) / 256), 256, 0, stream>>>(accum, zn);
    sage_scatter_kernel<<<(int)((sthreads + 255) / 256), 256, 0, stream>>>(tmp, esrc, edst, accum, E);
    sage_gemm_kernel<<<gemmGrid, 256, 0, stream>>>(tmp, accum, deg, Ws1, Wn1, b1,
                                                   nullptr, out, nullptr, 0.5f, nBlocks, N);
}